// CRF_51702816309403
// MI455X (gfx1250) — compile-verified
//
#include <hip/hip_runtime.h>
#include <hip/hip_bf16.h>

#define VIT_B 128
#define VIT_T 2048
#define VIT_C 128
#define NEG_HUGE (-3.402823466e38f)

typedef int           vit_v4i __attribute__((ext_vector_type(4)));
typedef unsigned int  vit_v4u __attribute__((ext_vector_type(4)));
typedef int           vit_v8i __attribute__((ext_vector_type(8)));

// ---------------------------------------------------------------------------
// gfx1250 async-to-LDS path (guarded; falls back to global_prefetch_b8)
// ---------------------------------------------------------------------------
#if defined(__has_builtin)
#  if __has_builtin(__builtin_amdgcn_global_load_async_to_lds_b128) && \
      __has_builtin(__builtin_amdgcn_s_wait_asynccnt)
#    define VIT_HAVE_ASYNC 1
#  endif
#  if __has_builtin(__builtin_amdgcn_tensor_load_to_lds) && \
      __has_builtin(__builtin_amdgcn_s_wait_tensorcnt)
#    define VIT_HAVE_TDM 1
#  endif
#endif
#ifndef VIT_HAVE_ASYNC
#  define VIT_HAVE_ASYNC 0
#endif
#ifndef VIT_HAVE_TDM
#  define VIT_HAVE_TDM 0
#endif

#if VIT_HAVE_ASYNC
typedef __attribute__((address_space(1))) vit_v4i* vit_gp_t;  // global int4*
typedef __attribute__((address_space(3))) vit_v4i* vit_lp_t;  // LDS int4*

__device__ __forceinline__ void vit_async_copy_b128(const float* g, float* l) {
  // generic global pointer bit-pattern == addrspace(1); LDS offset == low 32
  // bits of the generic LDS address (aperture truncation, ISA 10.2).
  __builtin_amdgcn_global_load_async_to_lds_b128(
      (vit_gp_t)(unsigned long long)g,
      (vit_lp_t)(unsigned int)(unsigned long long)l,
      0, 0);
}
#endif

#if VIT_HAVE_TDM
// One-shot TDM DMA of the 128x128 f32 transitions tile into LDS (ISA ch.7/8).
__device__ __forceinline__ void vit_tdm_load_f32_128x128(const float* g, float* lds) {
  unsigned long long ga = (unsigned long long)g;
  unsigned int la = (unsigned int)(unsigned long long)lds;   // LDS byte offset
  vit_v4u g0;
  g0.x = 1u;                                    // count=1, user descriptor, no gather
  g0.y = la;                                    // lds_addr [63:32]
  g0.z = (unsigned int)ga;                      // global_addr [95:64]
  g0.w = (unsigned int)((ga >> 32) & 0x01FFFFFFull) | (2u << 30); // addr[56:32] | type=2
  vit_v8i g1;
  g1.s0 = (int)(2u << 16);                      // workgroup_mask=0, data_size=2 (4B)
  g1.s1 = (int)(128u << 16);                    // tensor_dim0=128  (bits[79:48] lo16)
  g1.s2 = (int)(128u << 16);                    // tensor_dim1=128  (bits[111:80] lo16)
  g1.s3 = (int)(128u << 16);                    // tile_dim0=128    (bits[127:112])
  g1.s4 = 128;                                  // tile_dim1=128, tile_dim2=0
  g1.s5 = 128;                                  // tensor_dim0_stride=128 (lo32)
  g1.s6 = (int)(16384u << 16);                  // tensor_dim1_stride=16384 (lo16)
  g1.s7 = 0;
  vit_v4i z4 = {0, 0, 0, 0};                    // groups 2/3 unused (2D tensor)
  vit_v8i z8 = {0, 0, 0, 0, 0, 0, 0, 0};        // extra SGPR group (6-arg form)
  __builtin_amdgcn_tensor_load_to_lds(g0, g1, z4, z4, z8, 0);
}
#endif

// Butterfly argmax reduce over a wave32, ties -> lowest index (matches jnp.argmax)
__device__ __forceinline__ void vit_argmax_reduce(float& m, int& idx) {
#pragma unroll
  for (int off = 16; off > 0; off >>= 1) {
    float om = __shfl_xor(m, off, 32);
    int   oi = __shfl_xor(idx, off, 32);
    bool g = (om > m) || (om == m && oi < idx);
    m   = g ? om : m;
    idx = g ? oi : idx;
  }
}

// ---------------------------------------------------------------------------
// Plan B forward: max-only (2 VALU ops per max-plus sum), stores every alpha
// row to global (argmax recomputed in backward).  One WG per batch element,
// 512 threads: c = tid&127 owns an output tag, h = tid>>7 owns 32 prevs whose
// transition column lives entirely in VGPRs.
// ---------------------------------------------------------------------------
__global__ __launch_bounds__(512) void viterbi_fwd_max(
    const float* __restrict__ pot, const float* __restrict__ trans,
    float* __restrict__ alphaWS)
{
  __shared__ __align__(16) float alpha_s[VIT_C];
  __shared__ float part_s[3 * VIT_C];
  const int tid = threadIdx.x;
  const int c = tid & (VIT_C - 1);
  const int h = tid >> 7;              // 0..3 prev-range split
  const int b = blockIdx.x;

  float tr[32];
#pragma unroll
  for (int i = 0; i < 32; ++i) tr[i] = trans[(h * 32 + i) * VIT_C + c];

  const float* potB = pot + (size_t)b * VIT_T * VIT_C;
  float* AWb = alphaWS + (size_t)b * VIT_T * VIT_C;

  if (h == 0) {                         // t = 0: alpha0 = potentials[:,0]
    float a0 = potB[c];
    alpha_s[c] = a0;
    AWb[c] = a0;
  }
  __syncthreads();

  for (int t = 1; t < VIT_T; ++t) {
    if (h == 3 && (c & 31) == 0 && (t + 8) < VIT_T)
      __builtin_prefetch(potB + (size_t)(t + 8) * VIT_C + c, 0, 0); // global_prefetch_b8
    float pv = 0.0f;
    if (h == 0) pv = potB[(size_t)t * VIT_C + c];    // issued early, used late

    const float4* a4 = ((const float4*)alpha_s) + h * 8;
    float m0 = NEG_HUGE, m1 = NEG_HUGE, m2 = NEG_HUGE, m3 = NEG_HUGE;
#pragma unroll
    for (int i = 0; i < 8; ++i) {
      float4 a = a4[i];                 // broadcast ds_load_b128
      m0 = fmaxf(m0, a.x + tr[4 * i + 0]);
      m1 = fmaxf(m1, a.y + tr[4 * i + 1]);
      m2 = fmaxf(m2, a.z + tr[4 * i + 2]);
      m3 = fmaxf(m3, a.w + tr[4 * i + 3]);
    }
    float m = fmaxf(fmaxf(m0, m1), fmaxf(m2, m3));
    if (h) part_s[(h - 1) * VIT_C + c] = m;
    __syncthreads();                    // alpha reads + partials complete
    if (h == 0) {
      m = fmaxf(m, fmaxf(part_s[c], fmaxf(part_s[VIT_C + c], part_s[2 * VIT_C + c])));
      float an = m + pv;
      alpha_s[c] = an;
      AWb[(size_t)t * VIT_C + c] = an;  // coalesced 512B row store
    }
    __syncthreads();
  }
}

// ---------------------------------------------------------------------------
// Plan B backward: one wave per batch element; transitions DMA'd by the TDM
// into LDS (or gathered) then transposed; alpha rows streamed ahead of the
// pointer chase (async-to-LDS ring, or global_prefetch_b8 fallback).
// ---------------------------------------------------------------------------
__global__ __launch_bounds__(128) void viterbi_bwd_alpha(
    const float* __restrict__ alphaWS, const float* __restrict__ trans,
    float* __restrict__ out)
{
  __shared__ __align__(16) float Ttr[VIT_C * VIT_C];   // 64KB: Ttr[c][prev] = trans[prev][c]
#if VIT_HAVE_TDM
  __shared__ __align__(16) float Tstage[VIT_C * VIT_C]; // 64KB row-major TDM landing pad
#endif
#if VIT_HAVE_ASYNC
  __shared__ __align__(16) float aring[16 * VIT_C];    // 8KB ring of alpha rows
#endif
  const int tid = threadIdx.x;
  const int b = blockIdx.x;

#if VIT_HAVE_TDM
  if (tid < 32) {                                      // one wave issues the DMA
    vit_tdm_load_f32_128x128(trans, &Tstage[0]);
    __builtin_amdgcn_s_wait_tensorcnt(0);
  }
  __syncthreads();
  for (int k = tid; k < VIT_C * VIT_C; k += 128) {     // LDS->LDS transpose
    int cc = k >> 7, prev = k & (VIT_C - 1);
    Ttr[k] = Tstage[prev * VIT_C + cc];
  }
#else
  for (int k = tid; k < VIT_C * VIT_C; k += 128) {     // coalesced LDS writes
    int cc = k >> 7, prev = k & (VIT_C - 1);
    Ttr[k] = trans[prev * VIT_C + cc];
  }
#endif
  __syncthreads();
  if (tid >= 32) return;                               // one wave walks
  const int lane = tid;
  const float* Ab = alphaWS + (size_t)b * VIT_T * VIT_C;
  float* ob = out + (size_t)b * VIT_T;

#if VIT_HAVE_ASYNC
#pragma unroll
  for (int k = 0; k < 12; ++k) {                       // prime 12 rows in flight
    int row = VIT_T - 1 - k;
    vit_async_copy_b128(Ab + (size_t)row * VIT_C + 4 * lane,
                        &aring[(row & 15) * VIT_C + 4 * lane]);
  }
#else
  for (int k = 0; k < 16; ++k)
    if ((lane & 7) == 0)
      __builtin_prefetch(Ab + (size_t)(VIT_T - 1 - k) * VIT_C + (lane >> 3) * 32, 0, 0);
#endif

  auto loadrow = [&](int row) -> float4 {
#if VIT_HAVE_ASYNC
    if (row >= 12) __builtin_amdgcn_s_wait_asynccnt(11);  // retire oldest (in-order)
    else           __builtin_amdgcn_s_wait_asynccnt(0);   // tail drain
    float4 a = *(const float4*)&aring[(row & 15) * VIT_C + 4 * lane];
    int nrow = row - 12;
    if (nrow >= 0)
      vit_async_copy_b128(Ab + (size_t)nrow * VIT_C + 4 * lane,
                          &aring[(nrow & 15) * VIT_C + 4 * lane]);
    return a;
#else
    int p = row - 16;
    if (p >= 0 && (lane & 7) == 0)
      __builtin_prefetch(Ab + (size_t)p * VIT_C + (lane >> 3) * 32, 0, 0);
    return *(const float4*)(Ab + (size_t)row * VIT_C + 4 * lane);
#endif
  };

  // last_tag = argmax_c alpha[T-1]
  float4 a = loadrow(VIT_T - 1);
  float m = a.x; int idx = 4 * lane;
  if (a.y > m) { m = a.y; idx = 4 * lane + 1; }
  if (a.z > m) { m = a.z; idx = 4 * lane + 2; }
  if (a.w > m) { m = a.w; idx = 4 * lane + 3; }
  vit_argmax_reduce(m, idx);
  int tag = idx;
  if (lane == 0) ob[VIT_T - 1] = (float)tag;

  for (int t = VIT_T - 1; t >= 1; --t) {
    const int row = t - 1;
    float4 av = loadrow(row);
    float4 tv = *(const float4*)&Ttr[tag * VIT_C + 4 * lane];
    float s = av.x + tv.x; m = s; idx = 4 * lane;
    s = av.y + tv.y; if (s > m) { m = s; idx = 4 * lane + 1; }
    s = av.z + tv.z; if (s > m) { m = s; idx = 4 * lane + 2; }
    s = av.w + tv.w; if (s > m) { m = s; idx = 4 * lane + 3; }
    vit_argmax_reduce(m, idx);
    tag = idx;
    if (lane == 0) ob[row] = (float)tag;
  }
}

// ---------------------------------------------------------------------------
// Plan A (fallback when d_ws < 128MB): forward with uint8 backpointers (33.5MB)
// ---------------------------------------------------------------------------
__global__ __launch_bounds__(512) void viterbi_fwd_bp(
    const float* __restrict__ pot, const float* __restrict__ trans,
    unsigned char* __restrict__ bp, float* __restrict__ fa)
{
  __shared__ __align__(16) float alpha_s[VIT_C];
  __shared__ float part_m[3 * VIT_C];
  __shared__ int   part_i[3 * VIT_C];
  const int tid = threadIdx.x;
  const int c = tid & (VIT_C - 1);
  const int h = tid >> 7;
  const int b = blockIdx.x;

  float tr[32];
#pragma unroll
  for (int i = 0; i < 32; ++i) tr[i] = trans[(h * 32 + i) * VIT_C + c];

  const float* potB = pot + (size_t)b * VIT_T * VIT_C;
  unsigned char* bpB = bp + (size_t)b * VIT_T * VIT_C;
  if (h == 0) alpha_s[c] = potB[c];
  __syncthreads();

  for (int t = 1; t < VIT_T; ++t) {
    if (h == 3 && (c & 31) == 0 && (t + 8) < VIT_T)
      __builtin_prefetch(potB + (size_t)(t + 8) * VIT_C + c, 0, 0);
    float pv = 0.0f;
    if (h == 0) pv = potB[(size_t)t * VIT_C + c];

    const float4* a4 = ((const float4*)alpha_s) + h * 8;
    float m0 = NEG_HUGE, m1 = NEG_HUGE, m2 = NEG_HUGE, m3 = NEG_HUGE;
    int i0 = 0, i1 = 0, i2 = 0, i3 = 0;
#pragma unroll
    for (int i = 0; i < 8; ++i) {
      float4 aa = a4[i];
      const int base = h * 32 + 4 * i;
      float s;
      s = aa.x + tr[4 * i + 0]; if (s > m0) { m0 = s; i0 = base; }
      s = aa.y + tr[4 * i + 1]; if (s > m1) { m1 = s; i1 = base + 1; }
      s = aa.z + tr[4 * i + 2]; if (s > m2) { m2 = s; i2 = base + 2; }
      s = aa.w + tr[4 * i + 3]; if (s > m3) { m3 = s; i3 = base + 3; }
    }
    float m = m0; int idx = i0;        // combine chains, ties -> lowest prev
    if (m1 > m || (m1 == m && i1 < idx)) { m = m1; idx = i1; }
    if (m2 > m || (m2 == m && i2 < idx)) { m = m2; idx = i2; }
    if (m3 > m || (m3 == m && i3 < idx)) { m = m3; idx = i3; }
    if (h) { part_m[(h - 1) * VIT_C + c] = m; part_i[(h - 1) * VIT_C + c] = idx; }
    __syncthreads();
    if (h == 0) {
#pragma unroll
      for (int k = 0; k < 3; ++k) {    // strict > keeps lower-h = lower prev on tie
        float pm = part_m[k * VIT_C + c];
        if (pm > m) { m = pm; idx = part_i[k * VIT_C + c]; }
      }
      alpha_s[c] = m + pv;
      bpB[(size_t)t * VIT_C + c] = (unsigned char)idx;
    }
    __syncthreads();
  }
  if (h == 0) fa[b * VIT_C + c] = alpha_s[c];
}

__global__ __launch_bounds__(32) void viterbi_bwd_chase(
    const unsigned char* __restrict__ bp, const float* __restrict__ fa,
    float* __restrict__ out)
{
  const int b = blockIdx.x;
  const int lane = threadIdx.x;
  const float* fab = fa + b * VIT_C;
  float4 a = *(const float4*)(fab + 4 * lane);
  float m = a.x; int idx = 4 * lane;
  if (a.y > m) { m = a.y; idx = 4 * lane + 1; }
  if (a.z > m) { m = a.z; idx = 4 * lane + 2; }
  if (a.w > m) { m = a.w; idx = 4 * lane + 3; }
  vit_argmax_reduce(m, idx);
  int tag = idx;
  const unsigned char* bpB = bp + (size_t)b * VIT_T * VIT_C;
  float* ob = out + (size_t)b * VIT_T;
  if (lane == 0) ob[VIT_T - 1] = (float)tag;
  for (int t = VIT_T - 1; t >= 1; --t) {
    if (lane < 2 && t >= 17)
      __builtin_prefetch(bpB + (size_t)(t - 16) * VIT_C + lane * 64, 0, 0);
    if (lane == 0) {
      tag = bpB[(size_t)t * VIT_C + tag];
      ob[t - 1] = (float)tag;
    }
  }
}

// ---------------------------------------------------------------------------
extern "C" void kernel_launch(void* const* d_in, const int* in_sizes, int n_in,
                              void* d_out, int out_size, void* d_ws, size_t ws_size,
                              hipStream_t stream) {
  (void)in_sizes; (void)n_in; (void)out_size;
  const float* pot = (const float*)d_in[0];     // [B,T,C] f32
  const float* trans = (const float*)d_in[1];   // [C,C]   f32
  float* out = (float*)d_out;                   // [B,T]   f32 tags

  const size_t needB = (size_t)VIT_B * VIT_T * VIT_C * sizeof(float); // 128MB
  if (ws_size >= needB) {
    float* alphaWS = (float*)d_ws;
    viterbi_fwd_max<<<VIT_B, 512, 0, stream>>>(pot, trans, alphaWS);
    viterbi_bwd_alpha<<<VIT_B, 128, 0, stream>>>(alphaWS, trans, out);
  } else {
    unsigned char* bpW = (unsigned char*)d_ws;            // 33.5MB
    float* fa = (float*)(bpW + (size_t)VIT_B * VIT_T * VIT_C);
    viterbi_fwd_bp<<<VIT_B, 512, 0, stream>>>(pot, trans, bpW, fa);
    viterbi_bwd_chase<<<VIT_B, 32, 0, stream>>>(bpW, fa, out);
  }
}